// YOLOV3Loss_33457795236259
// MI455X (gfx1250) — compile-verified
//
#include <hip/hip_runtime.h>
#include <hip/hip_bf16.h>
#include <math.h>

// ---------------------------------------------------------------------------
// YOLOv3 loss for gfx1250 (MI455X). Bandwidth-bound single pass over pred.
// CDNA5-specific: V_WMMA_F32_16X16X4_F32 for the wave reduction
// (A = ones -> every lane's c[0] = val[lane%16] + val[lane%16+16]),
// ds_swizzle xor-reduction, global_prefetch_b8.
// Division-free IoU ignore test: max_t iou <= 0.7  <=>  all_t inter <= 0.7*(union+eps).
// ---------------------------------------------------------------------------

typedef __attribute__((ext_vector_type(2))) float v2f;
typedef __attribute__((ext_vector_type(8))) float v8f;

#define B_ANC 3
#define NCLS  20
#define TMAX  64
#define IGNORE_THRESH 0.7f

__constant__ float c_anchors[9][2] = {
    {10.f, 13.f}, {16.f, 30.f}, {33.f, 23.f},
    {30.f, 61.f}, {62.f, 45.f}, {59.f, 119.f},
    {116.f, 90.f}, {156.f, 198.f}, {373.f, 326.f}};

__device__ __forceinline__ float sigmoidf_(float x) {
    return 1.0f / (1.0f + expf(-x));
}

__device__ __forceinline__ float swz_add_(float s, float o) {
    // placeholder (not used) -- kept minimal
    return s + o;
}

__global__ void yolo_zero_kernel(float* out, int n) {
    int i = blockIdx.x * blockDim.x + threadIdx.x;
    if (i < n) out[i] = 0.0f;
}

// One block per (cell-chunk, n). blockDim.x == 256 (8 waves of 32).
__global__ void yolo_scale_loss(const float* __restrict__ pred,
                                const float* __restrict__ targets,
                                const int* __restrict__ img_size_p,
                                float* __restrict__ out,
                                float invN, int T, int g, int sidx) {
    // Per-target metadata (precomputed corners/areas for the hot loop).
    __shared__ float s_bx1[TMAX], s_bx2[TMAX], s_by1[TMAX], s_by2[TMAX];
    __shared__ float s_areab[TMAX];
    __shared__ float s_enc[TMAX][4];
    __shared__ float s_scale4[TMAX];
    __shared__ int   s_key[TMAX], s_tcls[TMAX];
    __shared__ float s_wsum[8];

    const int   n    = blockIdx.y;
    const float img  = (float)(*img_size_p);
    const float rimg = 1.0f / img;
    const float rg   = 1.0f / (float)g;

    // ---- cooperative per-target metadata for batch image n ----
    const int t = threadIdx.x;
    if (t < T && t < TMAX) {
        const float* tg = targets + ((size_t)n * T + t) * 5;
        const float cls = tg[0], x = tg[1], y = tg[2], w = tg[3], h = tg[4];

        // valid = cumprod over leading targets of (x != 0)
        int valid = 1;
        for (int tt = 0; tt <= t; ++tt)
            valid &= (targets[((size_t)n * T + tt) * 5 + 1] != 0.0f) ? 1 : 0;

        // anchor argmax (boxes co-centered -> inter = min(w,aw)*min(h,ah))
        float best = -1.0f;
        int   k    = 0;
        for (int a = 0; a < 9; ++a) {
            const float aw = c_anchors[a][0] * rimg;
            const float ah = c_anchors[a][1] * rimg;
            const float inter = fminf(w, aw) * fminf(h, ah);
            const float uni   = w * h + aw * ah - inter;
            const float iou   = inter / (uni + 1e-16f);
            if (iou > best) { best = iou; k = a; }
        }
        const int matched = valid && ((k / B_ANC) == sidx);
        const int local   = k % B_ANC;

        const float fii = fminf(fmaxf(floorf(x * (float)g), 0.0f), (float)(g - 1));
        const float fjj = fminf(fmaxf(floorf(y * (float)g), 0.0f), (float)(g - 1));
        const int ii = (int)fii, jj = (int)fjj;

        s_enc[t][0] = x * (float)g - (float)ii;
        s_enc[t][1] = y * (float)g - (float)jj;
        s_enc[t][2] = logf(fmaxf(w, 1e-12f) * img / c_anchors[k][0]);
        s_enc[t][3] = logf(fmaxf(h, 1e-12f) * img / c_anchors[k][1]);
        s_scale4[t] = 2.0f - w * h;
        s_tcls[t]   = (int)cls;
        s_key[t]    = matched ? ((local * g + jj) * g + ii) : -1;

        if (valid) {
            s_bx1[t] = x - w * 0.5f;  s_bx2[t] = x + w * 0.5f;
            s_by1[t] = y - h * 0.5f;  s_by2[t] = y + h * 0.5f;
            s_areab[t] = w * h;
        } else {
            // Sentinel: forces inter = 0 so the ignore test trivially passes
            // (matches reference's ious := -1 for invalid targets).
            s_bx1[t] =  1e30f;  s_bx2[t] = -1e30f;
            s_by1[t] =  1e30f;  s_by2[t] = -1e30f;
            s_areab[t] = 0.0f;
        }
    }
    __syncthreads();

    // ---- per-cell loss ----
    const int gg    = g * g;
    const int total = B_ANC * gg;
    const int cell  = blockIdx.x * blockDim.x + threadIdx.x;
    float partial   = 0.0f;

    if (cell < total) {
        const int b   = cell / gg;
        const int rem = cell - b * gg;
        const int j   = rem / g;
        const int i   = rem - j * g;

        const size_t st   = (size_t)gg;
        const size_t base = (((size_t)n * (B_ANC * (NCLS + 5)) +
                              (size_t)b * (NCLS + 5)) * g + (size_t)j) * g + i;

        __builtin_prefetch(pred + base, 0, 1);  // global_prefetch_b8

        const float tx    = pred[base];
        const float ty    = pred[base + st];
        const float tw    = pred[base + 2 * st];
        const float th    = pred[base + 3 * st];
        const float tconf = pred[base + 4 * st];

        const float sx    = sigmoidf_(tx);
        const float sy    = sigmoidf_(ty);
        const float sconf = sigmoidf_(tconf);

        const float aw = c_anchors[sidx * B_ANC + b][0];
        const float ah = c_anchors[sidx * B_ANC + b][1];
        const float px = (sx + (float)i) * rg;
        const float py = (sy + (float)j) * rg;
        const float pw = expf(tw) * aw * rimg;
        const float ph = expf(th) * ah * rimg;

        const float ax1 = px - pw * 0.5f, ax2 = px + pw * 0.5f;
        const float ay1 = py - ph * 0.5f, ay2 = py + ph * 0.5f;
        const float area_a = pw * ph;

        int  noobj = 1;   // all valid targets have iou <= 0.7
        int  mt    = -1;  // last matching target wins (duplicate scatter indices)
        const int mykey = (b * g + j) * g + i;

        for (int tt = 0; tt < T; ++tt) {
            const float iw = fmaxf(fminf(ax2, s_bx2[tt]) - fmaxf(ax1, s_bx1[tt]), 0.0f);
            const float ih = fmaxf(fminf(ay2, s_by2[tt]) - fmaxf(ay1, s_by1[tt]), 0.0f);
            const float inter = iw * ih;
            const float uni   = area_a + s_areab[tt] - inter;
            // iou <= 0.7  <=>  inter <= 0.7*(union + eps)   (division-free)
            noobj &= (inter <= IGNORE_THRESH * (uni + 1e-16f)) ? 1 : 0;
            if (s_key[tt] == mykey) mt = tt;
        }

        if (mt >= 0) {  // obj cell (noobj forcibly cleared at these indices)
            float d = sconf - 1.0f;
            partial += d * d;

            const int tc = s_tcls[mt];
            for (int c = 0; c < NCLS; ++c) {
                const float pc = sigmoidf_(pred[base + (size_t)(5 + c) * st]);
                const float dv = pc - ((c == tc) ? 1.0f : 0.0f);
                partial += dv * dv;
            }
            const float s4 = s_scale4[mt];
            const float d0 = (sx - s_enc[mt][0]) * s4;
            const float d1 = (sy - s_enc[mt][1]) * s4;
            const float d2 = (tw - s_enc[mt][2]) * s4;
            const float d3 = (th - s_enc[mt][3]) * s4;
            partial += d0 * d0 + d1 * d1 + d2 * d2 + d3 * d3;
        } else if (noobj) {
            partial += sconf * sconf;
        }
        partial *= invN;
    }

    // ---- wave reduction via WMMA (convergent: EXEC all-ones here) ----
    // A = ones(16x4); B vgpr0 = per-lane partial, vgpr1 = 0.
    // => every lane's c[0] = partial[lane%16] + partial[lane%16 + 16].
    v2f av; av.x = 1.0f;    av.y = 1.0f;
    v2f bv; bv.x = partial; bv.y = 0.0f;
    v8f cacc = {};
    cacc = __builtin_amdgcn_wmma_f32_16x16x4_f32(
        /*neg_a=*/false, av, /*neg_b=*/false, bv,
        /*c_mod=*/(short)0, cacc, /*reuse_a=*/false, /*reuse_b=*/false);

    // 16-way xor reduction with ds_swizzle (group-of-32, xor_mask in [14:10]).
    float s = cacc[0];
    s += __int_as_float(__builtin_amdgcn_ds_swizzle(__float_as_int(s), 0x041F)); // xor 1
    s += __int_as_float(__builtin_amdgcn_ds_swizzle(__float_as_int(s), 0x081F)); // xor 2
    s += __int_as_float(__builtin_amdgcn_ds_swizzle(__float_as_int(s), 0x101F)); // xor 4
    s += __int_as_float(__builtin_amdgcn_ds_swizzle(__float_as_int(s), 0x201F)); // xor 8

    const int lane = threadIdx.x & 31;
    const int wave = threadIdx.x >> 5;
    if (lane == 0) s_wsum[wave] = s;
    __syncthreads();
    if (threadIdx.x == 0) {
        float tot = 0.0f;
        const int nw = (blockDim.x + 31) >> 5;
        for (int wv = 0; wv < nw; ++wv) tot += s_wsum[wv];
        atomicAdd(out, tot);
    }
}

extern "C" void kernel_launch(void* const* d_in, const int* in_sizes, int n_in,
                              void* d_out, int out_size, void* d_ws, size_t ws_size,
                              hipStream_t stream) {
    const float* preds[3] = {(const float*)d_in[0],
                             (const float*)d_in[1],
                             (const float*)d_in[2]};
    const float* targets  = (const float*)d_in[3];
    const int*   img_size = (const int*)d_in[4];
    float*       out      = (float*)d_out;

    // Derive N, T from input sizes (reference: pred0 = N x 75 x 19 x 19).
    const int N = in_sizes[0] / (B_ANC * (NCLS + 5) * 19 * 19);
    const int T = in_sizes[3] / (N * 5);
    const float invN = 1.0f / (float)N;

    yolo_zero_kernel<<<1, 64, 0, stream>>>(out, out_size);

    const int gs[3] = {19, 38, 76};
    for (int s = 0; s < 3; ++s) {
        const int g     = gs[s];
        const int cells = B_ANC * g * g;
        dim3 grid((cells + 255) / 256, N);
        yolo_scale_loss<<<grid, 256, 0, stream>>>(preds[s], targets, img_size,
                                                  out, invN, T, g, s);
    }
}